// CostVolumeDisp_56745107914800
// MI455X (gfx1250) — compile-verified
//
#include <hip/hip_runtime.h>

// CDNA5 / gfx1250, wave32. 1-D horizontal cost volume (B,C,H,W)x(B,C,H,W)
// -> (B, 9, H, W) as a 9-wide diagonal band of the W x W Gram matrix per
// (b,h), computed with V_WMMA_F32_16X16X4_F32 (full f32 precision).
// HBM-bound at ~2.2 FLOP/B -> target is pure saddr-form b32 loads + WMMA.

typedef __attribute__((ext_vector_type(2))) float v2f;
typedef __attribute__((ext_vector_type(8))) float v8f;

// Compile-time problem shape (from the reference harness).
constexpr int B = 4, C = 128, H = 192, W = 640;
constexpr int DMAX = 4, NSHIFT = 9;         // 2*DMAX+1 shifts
constexpr int WBLK = W / 16;                // 40 column blocks
constexpr unsigned CH = (unsigned)(H * W);  // 122880 elems between channels

__global__ __launch_bounds__(256) void costvol_wmma_f32(
    const float* __restrict__ f1, const float* __restrict__ f2,
    float* __restrict__ out)
{
    const int lane = threadIdx.x & 31;
    const int wave = threadIdx.x >> 5;

    // One wave per (b, h, 16-wide w block). Grid is sized exactly, so no
    // guard; readfirstlane keeps the tile decomposition on the SALU.
    const int tile = __builtin_amdgcn_readfirstlane((int)(blockIdx.x * 8 + wave));

    const int wblk = tile % WBLK;
    const int h    = (tile / WBLK) % H;
    const int b    = tile / (WBLK * H);
    const int w0   = wblk << 4;

    const int laneM  = lane & 15;   // M (A rows / D cols) or N (B cols)
    const int laneHi = lane >> 4;   // 0 -> K=0,1 ; 1 -> K=2,3
    const int kOff   = laneHi << 1;

    // Uniform row base in elements (max ~62.9M, fits u32).
    const unsigned rowBase = (unsigned)(((b * C) * H + h) * W);

    // A rows always in range (W % 16 == 0). B columns clamped; the garbage
    // from clamped loads only reaches band entries we zero at store time.
    const int wA  = w0 + laneM;
    const int wB0 = w0 - DMAX + laneM;
    const int wB1 = wB0 + 16;
    const int wB0c = wB0 < 0 ? 0 : (wB0 >= W ? W - 1 : wB0);
    const int wB1c = wB1 < 0 ? 0 : (wB1 >= W ? W - 1 : wB1);

    unsigned offA  = rowBase + (unsigned)kOff * CH + (unsigned)wA;
    unsigned offB0 = rowBase + (unsigned)kOff * CH + (unsigned)wB0c;
    unsigned offB1 = rowBase + (unsigned)kOff * CH + (unsigned)wB1c;

    v8f acc0 = {};
    v8f acc1 = {};

#pragma unroll 4
    for (int k = 0; k < C; k += 4) {
        v2f a, b0, b1;
        a.x  = f1[offA];
        a.y  = f1[offA + CH];
        b0.x = f2[offB0];
        b0.y = f2[offB0 + CH];
        b1.x = f2[offB1];
        b1.y = f2[offB1 + CH];

        // D = A(16x4 f32) x B(4x16 f32) + C(16x16 f32)
        acc0 = __builtin_amdgcn_wmma_f32_16x16x4_f32(
            false, a, false, b0, (short)0, acc0, false, false);
        acc1 = __builtin_amdgcn_wmma_f32_16x16x4_f32(
            false, a, false, b1, (short)0, acc1, false, false);

        offA  += 4u * CH;
        offB0 += 4u * CH;
        offB1 += 4u * CH;
    }

    // Extract the 9-wide band, scale by 1/C, zero the padded-region entries.
    constexpr float scale = 1.0f / (float)C;
    const unsigned outRow = (unsigned)((b * NSHIFT * H + h) * W);

#pragma unroll
    for (int r = 0; r < 8; ++r) {
        const int M = r + (laneHi << 3);   // D row within tile
        const int w = w0 + M;

        // tile0: wcol = w0 - 4 + N  ->  j = N - M
        const int j0 = laneM - M;
        if (j0 >= 0 && j0 < NSHIFT) {
            const int wcol = w + j0 - DMAX;
            const float v = (wcol >= 0 && wcol < W) ? acc0[r] * scale : 0.0f;
            out[outRow + (unsigned)(j0 * H * W) + (unsigned)w] = v;
        }
        // tile1: wcol = w0 + 12 + N ->  j = N + 16 - M
        const int j1 = laneM + 16 - M;
        if (j1 >= 0 && j1 < NSHIFT) {
            const int wcol = w + j1 - DMAX;
            const float v = (wcol >= 0 && wcol < W) ? acc1[r] * scale : 0.0f;
            out[outRow + (unsigned)(j1 * H * W) + (unsigned)w] = v;
        }
    }
}

extern "C" void kernel_launch(void* const* d_in, const int* in_sizes, int n_in,
                              void* d_out, int out_size, void* d_ws, size_t ws_size,
                              hipStream_t stream) {
    const float* f1 = (const float*)d_in[0];
    const float* f2 = (const float*)d_in[1];
    float* out = (float*)d_out;
    (void)in_sizes; (void)n_in; (void)out_size; (void)d_ws; (void)ws_size;

    constexpr int ntiles = B * H * WBLK;      // 30720 (divisible by 8)
    constexpr int wavesPerBlock = 8;          // 256 threads = 8 wave32
    constexpr int blocks = ntiles / wavesPerBlock;  // 3840

    costvol_wmma_f32<<<blocks, wavesPerBlock * 32, 0, stream>>>(f1, f2, out);
}